// SINE_57329223467098
// MI455X (gfx1250) — compile-verified
//
#include <hip/hip_runtime.h>

// Problem constants (from reference: Q=200, C=1, H=W=800, TOPK=100)
#define Q             200
#define TOPK          100
#define HW            640000     // 800*800 floats per mask
#define F4_PER_BLOCK  1280       // 256 threads * 5 float4 each
#define BLOCKS_PER_MASK 125      // 160000 float4 / 1280
#define PART_STRIDE   128        // padded partial stride per mask

// d_out layout (floats): pm_bin[100*640000] | final_scores[100] | labels[100] | ids[100] | valid[100]
#define OUT_SCORES 64000000
#define OUT_LABELS 64000100
#define OUT_IDS    64000200
#define OUT_VALID  64000300

typedef __attribute__((ext_vector_type(2)))  float v2f;
typedef __attribute__((ext_vector_type(4)))  float v4f;   // native clang vector: OK for nontemporal builtins
typedef __attribute__((ext_vector_type(8)))  float v8f;

__device__ __forceinline__ float fast_sigmoid(float x) {
    return __fdividef(1.0f, 1.0f + __expf(-x));
}

// ---------------------------------------------------------------------------
// Kernel 1: scores = softmax(logits)[:, 0] = sigmoid(l0 - l1); stable top-100
// via O(Q^2) rank (descending, ties -> lower index, matching jax.lax.top_k).
// ---------------------------------------------------------------------------
__global__ __launch_bounds__(256) void topk_kernel(
    const float* __restrict__ logits, const int* __restrict__ labels,
    const int* __restrict__ ids, float* __restrict__ out,
    int* __restrict__ topk_idx, float* __restrict__ topk_score)
{
    __shared__ float sc[Q];
    const int t = threadIdx.x;
    if (t < Q) {
        const float l0 = logits[2 * t];
        const float l1 = logits[2 * t + 1];
        sc[t] = fast_sigmoid(l0 - l1);
    }
    __syncthreads();
    if (t < Q) {
        const float s = sc[t];
        int r = 0;
        for (int p = 0; p < Q; ++p) {
            const float sp = sc[p];
            r += (sp > s) || ((sp == s) && (p < t));
        }
        if (r < TOPK) {
            topk_idx[r]   = t;
            topk_score[r] = s;
            out[OUT_LABELS + r] = (float)labels[t];
            out[OUT_IDS + r]    = (float)ids[t];
        }
    }
}

// ---------------------------------------------------------------------------
// Kernel 2: stream one (mask, chunk): binarize + NT-store, accumulate
// sum(sigmoid*bin) and sum(bin). Streaming data has zero reuse (256 MB input
// > 192 MB L2, output never re-read) -> nontemporal loads/stores (TH=NT).
// Wave32 reduction via V_WMMA_F32_16X16X4_F32 with B = ones (exact f32);
// column-0 totals land in lanes 0 and 16. Deterministic: one partial per
// block written to a fixed slot (no atomics).
// ---------------------------------------------------------------------------
__global__ __launch_bounds__(256) void mask_kernel(
    const float* __restrict__ masks, float* __restrict__ out,
    const int* __restrict__ topk_idx,
    float* __restrict__ part_sig, float* __restrict__ part_bin)
{
    const int k  = blockIdx.y;     // output slot 0..99
    const int bx = blockIdx.x;     // chunk 0..124
    const int q  = topk_idx[k];    // source mask index

    const v4f* __restrict__ in = (const v4f*)(masks + (size_t)q * HW);
    v4f* __restrict__ op       = (v4f*)(out + (size_t)k * HW);

    int idx = bx * F4_PER_BLOCK + threadIdx.x;
    float acc_s = 0.0f, acc_b = 0.0f;
#pragma unroll
    for (int i = 0; i < 5; ++i, idx += 256) {
        const v4f x = __builtin_nontemporal_load(&in[idx]);
        v4f b;
        b.x = x.x > 0.0f ? 1.0f : 0.0f;
        b.y = x.y > 0.0f ? 1.0f : 0.0f;
        b.z = x.z > 0.0f ? 1.0f : 0.0f;
        b.w = x.w > 0.0f ? 1.0f : 0.0f;
        __builtin_nontemporal_store(b, &op[idx]);
        acc_s += b.x * fast_sigmoid(x.x) + b.y * fast_sigmoid(x.y)
               + b.z * fast_sigmoid(x.z) + b.w * fast_sigmoid(x.w);
        acc_b += b.x + b.y + b.z + b.w;
    }

    // ---- wave32 cross-lane reduction on the matrix pipe (EXEC is all 1s:
    //      no divergence above). D[m][n] = sum_k A[m][k] with B = ones;
    //      total of column 0 = (sum of 8 acc VGPRs in lane 0) + (lane 16).
    v8f cs = {};
    v8f cb = {};
    v2f a;    a.x = acc_s; a.y = 0.0f;
    v2f ones; ones.x = 1.0f; ones.y = 1.0f;
    cs = __builtin_amdgcn_wmma_f32_16x16x4_f32(false, a, false, ones,
                                               (short)0, cs, false, false);
    a.x = acc_b;
    cb = __builtin_amdgcn_wmma_f32_16x16x4_f32(false, a, false, ones,
                                               (short)0, cb, false, false);
    const float ts = cs[0]+cs[1]+cs[2]+cs[3]+cs[4]+cs[5]+cs[6]+cs[7];
    const float tb = cb[0]+cb[1]+cb[2]+cb[3]+cb[4]+cb[5]+cb[6]+cb[7];

    __shared__ float wsum_s[16];
    __shared__ float wsum_b[16];
    const int lane = threadIdx.x & 31;
    const int wave = threadIdx.x >> 5;
    if (lane == 0)  { wsum_s[2 * wave]     = ts; wsum_b[2 * wave]     = tb; }
    if (lane == 16) { wsum_s[2 * wave + 1] = ts; wsum_b[2 * wave + 1] = tb; }
    __syncthreads();
    if (threadIdx.x == 0) {
        float bs = 0.0f, bb = 0.0f;
#pragma unroll
        for (int i = 0; i < 16; ++i) { bs += wsum_s[i]; bb += wsum_b[i]; }
        part_sig[k * PART_STRIDE + bx] = bs;
        part_bin[k * PART_STRIDE + bx] = bb;
    }
}

// ---------------------------------------------------------------------------
// Kernel 3: ordered (deterministic) reduction of the 125 partials per mask,
// final_scores = topk_score * sum_sig / (sum_bin + 1e-6), valid = 1.
// ---------------------------------------------------------------------------
__global__ __launch_bounds__(128) void finalize_kernel(
    float* __restrict__ out, const float* __restrict__ topk_score,
    const float* __restrict__ part_sig, const float* __restrict__ part_bin)
{
    const int k = threadIdx.x;
    if (k < TOPK) {
        float ss = 0.0f, sb = 0.0f;
        for (int i = 0; i < BLOCKS_PER_MASK; ++i) {
            ss += part_sig[k * PART_STRIDE + i];
            sb += part_bin[k * PART_STRIDE + i];
        }
        const float ms = ss / (sb + 1e-6f);
        out[OUT_SCORES + k] = topk_score[k] * ms;
        out[OUT_VALID  + k] = 1.0f;
    }
}

// ---------------------------------------------------------------------------
extern "C" void kernel_launch(void* const* d_in, const int* in_sizes, int n_in,
                              void* d_out, int out_size, void* d_ws, size_t ws_size,
                              hipStream_t stream) {
    const float* masks  = (const float*)d_in[0];   // [200, 800, 800] f32
    const float* logits = (const float*)d_in[1];   // [200, 2] f32
    const int*   labels = (const int*)d_in[2];     // [200] int
    const int*   ids    = (const int*)d_in[3];     // [200] int
    float* out = (float*)d_out;

    // Workspace layout (~103 KB): topk_idx | topk_score | part_sig | part_bin
    char* ws = (char*)d_ws;
    int*   topk_idx   = (int*)(ws);
    float* topk_score = (float*)(ws + 512);
    float* part_sig   = (float*)(ws + 1024);
    float* part_bin   = (float*)(ws + 1024 + (size_t)TOPK * PART_STRIDE * sizeof(float));

    topk_kernel<<<1, 256, 0, stream>>>(logits, labels, ids, out,
                                       topk_idx, topk_score);
    mask_kernel<<<dim3(BLOCKS_PER_MASK, TOPK), 256, 0, stream>>>(
        masks, out, topk_idx, part_sig, part_bin);
    finalize_kernel<<<1, 128, 0, stream>>>(out, topk_score, part_sig, part_bin);
}